// HMN_22136261444044
// MI455X (gfx1250) — compile-verified
//
#include <hip/hip_runtime.h>

typedef float v2f __attribute__((ext_vector_type(2)));
typedef float v8f __attribute__((ext_vector_type(8)));

#define VB 64      // batch
#define VS 128     // segments per sample
#define VW 32      // words per segment
#define VT 128     // law segments
#define VWL 32     // words per law segment
#define VD 200     // embedding dim
#define VH 400     // 2*D hidden

// workspace layout (float offsets)
#define OFF_SEG   0            // B*S*D = 1,638,400
#define OFF_LAW   1638400      // T*D   =    25,600
#define OFF_FEAT  1664000      // B*H   =    25,600
#define OFF_H     1689600      // B*H   =    25,600
#define OFF_W2P   1715200      // H*16  =     6,400
#define OFF_LOG   1721600      // B*16  =     1,024

// ---------------------------------------------------------------------------
// Per-segment embedding sums (skip masked segments entirely -> saves ~50% of
// the 210 MB gather, the dominant cost on the 23.3 TB/s roofline).
// ---------------------------------------------------------------------------
__global__ void k_text_seg(const int* __restrict__ ids, const int* __restrict__ lens,
                           const float* __restrict__ emb, float* __restrict__ seg) {
    const int s = blockIdx.x, b = blockIdx.y;
    if (s >= lens[b]) return;                 // uniform per block: no traffic for masked segs
    __shared__ int wid[VW];
    const int tid = threadIdx.x;
    if (tid < VW) wid[tid] = ids[(b * VS + s) * VW + tid];
    __syncthreads();
    if (tid >= VD) return;
    float acc = 0.f;
#pragma unroll 4
    for (int w = 0; w < VW; ++w)
        acc += emb[(size_t)wid[w] * VD + tid];
    seg[(size_t)(b * VS + s) * VD + tid] = acc;
}

__global__ void k_law_seg(const int* __restrict__ ids, const float* __restrict__ emb,
                          float* __restrict__ lawS) {
    const int t = blockIdx.x;
    __shared__ int wid[VWL];
    const int tid = threadIdx.x;
    if (tid < VWL) wid[tid] = ids[t * VWL + tid];
    __syncthreads();
    if (tid >= VD) return;
    float acc = 0.f;
#pragma unroll 4
    for (int w = 0; w < VWL; ++w)
        acc += emb[(size_t)wid[w] * VD + tid];
    lawS[t * VD + tid] = acc;
}

// ---------------------------------------------------------------------------
// Deterministic reductions. avg_alpha == 1/T and avg_beta == 1/S analytically
// (softmax sums to 1 along its axis), so the AoA block collapses to scales.
//   fact[b]  = (1+1/T) * (1/(S*W)) * sum_{s<len}  seg_sum[b,s,:]
//   label    = (1+1/S) * (1/(T*WL)) * sum_t law_seg[t,:]   (same for all b)
// Both are written straight into feat[b][0:200] / feat[b][200:400].
// ---------------------------------------------------------------------------
__global__ void k_fact_reduce(const float* __restrict__ seg, const int* __restrict__ lens,
                              float* __restrict__ feat) {
    const int b = blockIdx.x, d = threadIdx.x;
    if (d >= VD) return;
    const int len = lens[b];
    float acc = 0.f;
    for (int s = 0; s < len; ++s)
        acc += seg[(size_t)(b * VS + s) * VD + d];
    const float scale = (1.0f + 1.0f / (float)VT) / ((float)VS * (float)VW);
    feat[b * VH + d] = acc * scale;
}

__global__ void k_law_reduce(const float* __restrict__ lawS, float* __restrict__ feat) {
    const int b = blockIdx.x, d = threadIdx.x;
    if (d >= VD) return;
    float acc = 0.f;
    for (int t = 0; t < VT; ++t)
        acc += lawS[t * VD + d];
    const float scale = (1.0f + 1.0f / (float)VS) / ((float)VT * (float)VWL);
    feat[b * VH + VD + d] = acc * scale;
}

// Zero-pad W2 [400][6] -> [400][16] so the head GEMM can use a 16-wide tile.
__global__ void k_w2pad(const float* __restrict__ W2, float* __restrict__ w2p) {
    const int idx = blockIdx.x * 256 + threadIdx.x;
    if (idx >= VH * 16) return;
    const int k = idx >> 4, j = idx & 15;
    w2p[idx] = (j < 6) ? W2[k * 6 + j] : 0.f;
}

// ---------------------------------------------------------------------------
// WMMA fp32 GEMMs (V_WMMA_F32_16X16X4_F32).
// Fragment layout per CDNA5 ISA §7.12.2:
//   A 16x4 f32, 2 VGPRs: v0 = K0 (lanes 0-15) / K2 (lanes 16-31); v1 = K1/K3
//   B 4x16 f32, 2 VGPRs: symmetric (row K striped across lanes within a VGPR)
//   C/D 16x16 f32, 8 VGPRs: vi = row i (lanes 0-15) / row i+8 (lanes 16-31)
// ---------------------------------------------------------------------------
__device__ __forceinline__ v8f wmma_f32x4(v2f a, v2f b, v8f c) {
    return __builtin_amdgcn_wmma_f32_16x16x4_f32(
        /*neg_a=*/false, a, /*neg_b=*/false, b,
        /*c_mod=*/(short)0, c, /*reuse_a=*/false, /*reuse_b=*/false);
}

// h[64][400] = relu(feat[64][400] @ W1[400][400] + b1)
__global__ void k_gemm1(const float* __restrict__ feat, const float* __restrict__ W1,
                        const float* __restrict__ b1, float* __restrict__ h) {
    const int wave = blockIdx.x * (blockDim.x >> 5) + (threadIdx.x >> 5);
    if (wave >= (VB / 16) * (VH / 16)) return;   // 4 * 25 = 100 tiles, wave-uniform exit
    const int lane = threadIdx.x & 31;
    const int half = lane >> 4, l = lane & 15;
    const int tm = wave / (VH / 16), tn = wave % (VH / 16);
    const int row = tm * 16 + l;
    const int col = tn * 16 + l;
    v8f c = {};
    for (int k0 = 0; k0 < VH; k0 += 4) {
        v2f a, bf;
        a.x  = feat[row * VH + k0 + 2 * half];
        a.y  = feat[row * VH + k0 + 2 * half + 1];
        bf.x = W1[(k0 + 2 * half) * VH + col];
        bf.y = W1[(k0 + 2 * half + 1) * VH + col];
        c = wmma_f32x4(a, bf, c);
    }
    const float bias = b1[col];
#pragma unroll
    for (int i = 0; i < 8; ++i) {
        const int m = tm * 16 + i + 8 * half;
        const float v = c[i] + bias;
        h[m * VH + col] = v > 0.f ? v : 0.f;
    }
}

// logits[64][16] = h[64][400] @ W2p[400][16] + b2 (cols >= 6 unused)
__global__ void k_gemm2(const float* __restrict__ h, const float* __restrict__ w2p,
                        const float* __restrict__ b2, float* __restrict__ logits) {
    const int wave = threadIdx.x >> 5;           // 4 waves, one 16-row tile each
    const int lane = threadIdx.x & 31;
    const int half = lane >> 4, l = lane & 15;
    const int row = wave * 16 + l;
    const int col = l;
    v8f c = {};
    for (int k0 = 0; k0 < VH; k0 += 4) {
        v2f a, bf;
        a.x  = h[row * VH + k0 + 2 * half];
        a.y  = h[row * VH + k0 + 2 * half + 1];
        bf.x = w2p[(k0 + 2 * half) * 16 + col];
        bf.y = w2p[(k0 + 2 * half + 1) * 16 + col];
        c = wmma_f32x4(a, bf, c);
    }
    const float bias = (col < 6) ? b2[col] : 0.f;
#pragma unroll
    for (int i = 0; i < 8; ++i) {
        const int m = wave * 16 + i + 8 * half;
        logits[m * 16 + col] = c[i] + bias;
    }
}

// Softmax over 6 classes -> d_out[0:384]; avg_alpha == 1/T -> d_out[384:8576].
__global__ void k_final(const float* __restrict__ logits, float* __restrict__ out) {
    const int b = blockIdx.x, tid = threadIdx.x;
    if (tid == 0) {
        float l[6], mx = -1e30f;
#pragma unroll
        for (int j = 0; j < 6; ++j) { l[j] = logits[b * 16 + j]; mx = fmaxf(mx, l[j]); }
        float sum = 0.f;
#pragma unroll
        for (int j = 0; j < 6; ++j) { l[j] = __expf(l[j] - mx); sum += l[j]; }
        const float inv = 1.f / sum;
#pragma unroll
        for (int j = 0; j < 6; ++j) out[b * 6 + j] = l[j] * inv;
    }
    // avg_alpha[b, s] = mean over softmax axis = 1/T (block has 128 threads = S)
    out[VB * 6 + b * VS + tid] = 1.0f / (float)VT;
}

extern "C" void kernel_launch(void* const* d_in, const int* in_sizes, int n_in,
                              void* d_out, int out_size, void* d_ws, size_t ws_size,
                              hipStream_t stream) {
    const int*   text_ids  = (const int*)d_in[0];
    const int*   text_lens = (const int*)d_in[1];
    const int*   law_ids   = (const int*)d_in[2];
    const float* emb       = (const float*)d_in[3];
    const float* W1        = (const float*)d_in[4];
    const float* b1        = (const float*)d_in[5];
    const float* W2        = (const float*)d_in[6];
    const float* b2        = (const float*)d_in[7];
    float* ws  = (float*)d_ws;
    float* out = (float*)d_out;

    float* seg   = ws + OFF_SEG;
    float* lawS  = ws + OFF_LAW;
    float* feat  = ws + OFF_FEAT;
    float* hbuf  = ws + OFF_H;
    float* w2p   = ws + OFF_W2P;
    float* logit = ws + OFF_LOG;

    k_text_seg   <<<dim3(VS, VB), 256, 0, stream>>>(text_ids, text_lens, emb, seg);
    k_law_seg    <<<VT,           256, 0, stream>>>(law_ids, emb, lawS);
    k_w2pad      <<<(VH * 16 + 255) / 256, 256, 0, stream>>>(W2, w2p);
    k_fact_reduce<<<VB,           256, 0, stream>>>(seg, text_lens, feat);
    k_law_reduce <<<VB,           256, 0, stream>>>(lawS, feat);
    k_gemm1      <<<13,           256, 0, stream>>>(feat, W1, b1, hbuf);   // 100 waves
    k_gemm2      <<<1,            128, 0, stream>>>(hbuf, w2p, b2, logit); // 4 waves
    k_final      <<<VB,           128, 0, stream>>>(logit, out);
}